// NeuralTangentKernel_73993696576147
// MI455X (gfx1250) — compile-verified
//
#include <hip/hip_runtime.h>

// Neural Tangent Kernel, factorized:  out = J_te @ (J_tr^T @ c)
//  Phase A (train): forward + VJP with cotangent c  -> parameter gradient v
//  Phase B (test):  forward + JVP with tangent v    -> (128,4) output
// All big matmuls use V_WMMA_F32_16X16X4_F32 (fp32 in/out, full precision);
// problem is launch-latency bound (~60 MFLOP, <1.5 MB), so fp32 is free.

#define D_IN 64
#define HID  256
#define DOUT 4
#define NTR  128
#define NTE  128

typedef __attribute__((ext_vector_type(2))) float v2f;
typedef __attribute__((ext_vector_type(8))) float v8f;

// ---- f32 16x16x4 WMMA fragment loaders (per ISA 7.12.2 layouts) ----
// A 16x4 (MxK): lane r (=lane&15) holds row M=m0+r; half h (=lane>>4) holds K pair {2h, 2h+1}
__device__ __forceinline__ v2f ld_a(const float* A, int lda, int m0, int k0, int lane) {
  int r = lane & 15, h = lane >> 4;
  const float* p = A + (size_t)(m0 + r) * lda + (k0 + 2 * h);
  v2f v; v.x = p[0]; v.y = p[1]; return v;
}
// A^T fragment: logical A' = A^T, A stored (K x M) row-major with row length lda (== M)
__device__ __forceinline__ v2f ld_aT(const float* A, int lda, int m0, int k0, int lane) {
  int r = lane & 15, h = lane >> 4;
  const float* p = A + (size_t)(k0 + 2 * h) * lda + (m0 + r);
  v2f v; v.x = p[0]; v.y = p[lda]; return v;
}
// B 4x16 (KxN): VGPR0 row K=2h, VGPR1 row K=2h+1, col N=n0+r striped across lanes
__device__ __forceinline__ v2f ld_b(const float* B, int ldb, int k0, int n0, int lane) {
  int r = lane & 15, h = lane >> 4;
  const float* p = B + (size_t)(k0 + 2 * h) * ldb + (n0 + r);
  v2f v; v.x = p[0]; v.y = p[ldb]; return v;
}
// B^T fragment: logical B' = B^T, B stored (N x K) row-major with row length ldb (== K)
__device__ __forceinline__ v2f ld_bT(const float* B, int ldb, int k0, int n0, int lane) {
  int r = lane & 15, h = lane >> 4;
  const float* p = B + (size_t)(n0 + r) * ldb + (k0 + 2 * h);
  v2f v; v.x = p[0]; v.y = p[1]; return v;
}

__device__ __forceinline__ v8f wmma4(v2f a, v2f b, v8f c) {
  // 8 args: (neg_a, A, neg_b, B, c_mod, C, reuse_a, reuse_b)
  return __builtin_amdgcn_wmma_f32_16x16x4_f32(false, a, false, b, (short)0, c,
                                               false, false);
}

// H = tanh(A@B + bias); A: MxK, B: KxN, bias: N.  One wave per 16x16 tile.
__global__ void k_mm_bias_tanh(const float* __restrict__ A, const float* __restrict__ B,
                               const float* __restrict__ bias, float* __restrict__ H,
                               int M, int K, int N) {
  int lane = threadIdx.x;
  int tn = N >> 4;
  int m0 = (blockIdx.x / tn) << 4, n0 = (blockIdx.x % tn) << 4;
  v8f acc = {};
  for (int k0 = 0; k0 < K; k0 += 4)
    acc = wmma4(ld_a(A, K, m0, k0, lane), ld_b(B, N, k0, n0, lane), acc);
  int r = lane & 15, hh = lane >> 4;
  int col = n0 + r;
  float bv = bias[col];
#pragma unroll
  for (int v = 0; v < 8; v++) {
    int row = m0 + v + 8 * hh;
    H[(size_t)row * N + col] = tanhf(acc[v] + bv);
  }
}

// C = A^T @ B; A stored (K x M) row-major, B: KxN
__global__ void k_mm_at(const float* __restrict__ A, const float* __restrict__ B,
                        float* __restrict__ C, int M, int K, int N) {
  int lane = threadIdx.x;
  int tn = N >> 4;
  int m0 = (blockIdx.x / tn) << 4, n0 = (blockIdx.x % tn) << 4;
  v8f acc = {};
  for (int k0 = 0; k0 < K; k0 += 4)
    acc = wmma4(ld_aT(A, M, m0, k0, lane), ld_b(B, N, k0, n0, lane), acc);
  int r = lane & 15, hh = lane >> 4;
  int col = n0 + r;
#pragma unroll
  for (int v = 0; v < 8; v++) {
    int row = m0 + v + 8 * hh;
    C[(size_t)row * N + col] = acc[v];
  }
}

// Out = (A @ B^T) * (1 - Hp^2); A: MxK, B stored (N x K), Hp/Out: MxN
__global__ void k_mm_bt_dtanh(const float* __restrict__ A, const float* __restrict__ B,
                              const float* __restrict__ Hp, float* __restrict__ Out,
                              int M, int K, int N) {
  int lane = threadIdx.x;
  int tn = N >> 4;
  int m0 = (blockIdx.x / tn) << 4, n0 = (blockIdx.x % tn) << 4;
  v8f acc = {};
  for (int k0 = 0; k0 < K; k0 += 4)
    acc = wmma4(ld_a(A, K, m0, k0, lane), ld_bT(B, K, k0, n0, lane), acc);
  int r = lane & 15, hh = lane >> 4;
  int col = n0 + r;
#pragma unroll
  for (int v = 0; v < 8; v++) {
    int row = m0 + v + 8 * hh;
    float hp = Hp[(size_t)row * N + col];
    Out[(size_t)row * N + col] = acc[v] * (1.0f - hp * hp);
  }
}

// dZ2[b][j] = (sum_t c[b][t] * W3[j][t]) * (1 - H2[b][j]^2)   (K=4: plain VALU)
__global__ void k_dz2(const float* __restrict__ Ctr, const float* __restrict__ W3,
                      const float* __restrict__ H2, float* __restrict__ dZ2) {
  int i = blockIdx.x * blockDim.x + threadIdx.x;
  if (i >= NTR * HID) return;
  int b = i / HID, j = i % HID;
  float s = 0.f;
#pragma unroll
  for (int t = 0; t < DOUT; t++) s += Ctr[b * DOUT + t] * W3[j * DOUT + t];
  float h = H2[i];
  dZ2[i] = s * (1.0f - h * h);
}

// gW3[j][t] = sum_b H2[b][j] * c[b][t]
__global__ void k_gw3(const float* __restrict__ H2, const float* __restrict__ Ctr,
                      float* __restrict__ gW3) {
  int i = blockIdx.x * blockDim.x + threadIdx.x;
  if (i >= HID * DOUT) return;
  int j = i / DOUT, t = i % DOUT;
  float s = 0.f;
  for (int b = 0; b < NTR; b++) s += H2[b * HID + j] * Ctr[b * DOUT + t];
  gW3[i] = s;
}

// out[c] = sum_r A[r][c]
__global__ void k_colsum(const float* __restrict__ A, float* __restrict__ out,
                         int rows, int cols) {
  int c = blockIdx.x * blockDim.x + threadIdx.x;
  if (c >= cols) return;
  float s = 0.f;
  for (int r = 0; r < rows; r++) s += A[(size_t)r * cols + c];
  out[c] = s;
}

// Test layer 1 (JVP): Z1=Xte@W1+b1, T1=Xte@gW1+gb1 ; H1t=tanh(Z1), dH1t=T1*(1-H1t^2)
__global__ void k_test_l1(const float* __restrict__ Xte, const float* __restrict__ W1,
                          const float* __restrict__ b1, const float* __restrict__ gW1,
                          const float* __restrict__ gb1, float* __restrict__ H1t,
                          float* __restrict__ dH1t) {
  int lane = threadIdx.x;
  int tn = HID >> 4;
  int m0 = (blockIdx.x / tn) << 4, n0 = (blockIdx.x % tn) << 4;
  v8f accz = {}, acct = {};
  for (int k0 = 0; k0 < D_IN; k0 += 4) {
    v2f a = ld_a(Xte, D_IN, m0, k0, lane);
    accz = wmma4(a, ld_b(W1, HID, k0, n0, lane), accz);
    acct = wmma4(a, ld_b(gW1, HID, k0, n0, lane), acct);
  }
  int r = lane & 15, hh = lane >> 4;
  int col = n0 + r;
  float bz = b1[col], bt = gb1[col];
#pragma unroll
  for (int v = 0; v < 8; v++) {
    int row = m0 + v + 8 * hh;
    float h = tanhf(accz[v] + bz);
    H1t[(size_t)row * HID + col] = h;
    dH1t[(size_t)row * HID + col] = (acct[v] + bt) * (1.0f - h * h);
  }
}

// Test layer 2: Z2=H1t@W2+b2, T2=dH1t@W2 + H1t@gW2 + gb2 ; H2t=tanh, dH2t=T2*(1-H2t^2)
__global__ void k_test_l2(const float* __restrict__ H1t, const float* __restrict__ dH1t,
                          const float* __restrict__ W2, const float* __restrict__ b2,
                          const float* __restrict__ gW2, const float* __restrict__ gb2,
                          float* __restrict__ H2t, float* __restrict__ dH2t) {
  int lane = threadIdx.x;
  int tn = HID >> 4;
  int m0 = (blockIdx.x / tn) << 4, n0 = (blockIdx.x % tn) << 4;
  v8f accz = {}, acct = {};
  for (int k0 = 0; k0 < HID; k0 += 4) {
    v2f a1 = ld_a(H1t, HID, m0, k0, lane);
    v2f a2 = ld_a(dH1t, HID, m0, k0, lane);
    v2f bw = ld_b(W2, HID, k0, n0, lane);
    v2f bg = ld_b(gW2, HID, k0, n0, lane);
    accz = wmma4(a1, bw, accz);
    acct = wmma4(a2, bw, acct);
    acct = wmma4(a1, bg, acct);
  }
  int r = lane & 15, hh = lane >> 4;
  int col = n0 + r;
  float bz = b2[col], bt = gb2[col];
#pragma unroll
  for (int v = 0; v < 8; v++) {
    int row = m0 + v + 8 * hh;
    float h = tanhf(accz[v] + bz);
    H2t[(size_t)row * HID + col] = h;
    dH2t[(size_t)row * HID + col] = (acct[v] + bt) * (1.0f - h * h);
  }
}

// out[a][t] = gb3[t] + sum_j dH2t[a][j]*W3[j][t] + H2t[a][j]*gW3[j][t]
__global__ void k_test_out(const float* __restrict__ H2t, const float* __restrict__ dH2t,
                           const float* __restrict__ W3, const float* __restrict__ gW3,
                           const float* __restrict__ gb3, float* __restrict__ out) {
  int i = blockIdx.x * blockDim.x + threadIdx.x;
  if (i >= NTE * DOUT) return;
  int a = i / DOUT, t = i % DOUT;
  float s = gb3[t];
  for (int j = 0; j < HID; j++)
    s += dH2t[a * HID + j] * W3[j * DOUT + t] + H2t[a * HID + j] * gW3[j * DOUT + t];
  out[i] = s;
}

extern "C" void kernel_launch(void* const* d_in, const int* in_sizes, int n_in,
                              void* d_out, int out_size, void* d_ws, size_t ws_size,
                              hipStream_t stream) {
  (void)in_sizes; (void)n_in; (void)out_size; (void)ws_size;
  const float* Xte = (const float*)d_in[0];
  const float* Xtr = (const float*)d_in[1];
  const float* W1  = (const float*)d_in[2];
  const float* b1  = (const float*)d_in[3];
  const float* W2  = (const float*)d_in[4];
  const float* b2  = (const float*)d_in[5];
  const float* W3  = (const float*)d_in[6];
  // d_in[7] = b3: Jacobians do not depend on b3's value (affine), unused.
  const float* Ctr = (const float*)d_in[8];  // (128,4) row-major
  float* out = (float*)d_out;

  // workspace layout (floats); buffers reused across phases (~860 KB total)
  float* ws   = (float*)d_ws;
  float* BUF1 = ws;            // H1 (train), later H1t (test)
  float* BUF2 = ws + 32768;    // H2 (train), then dZ1, then H2t (test)
  float* BUF3 = ws + 65536;    // dZ2 (train), then dH1t (test)
  float* BUF4 = ws + 98304;    // dH2t (test)
  float* gW1  = ws + 131072;   // 64x256
  float* gW2  = ws + 147456;   // 256x256
  float* gW3  = ws + 212992;   // 256x4
  float* gb1  = ws + 214016;   // 256
  float* gb2  = ws + 214272;   // 256
  float* gb3  = ws + 214528;   // 4

  const int W = 32;  // one wave32 per 16x16 tile

  // ---- Phase A: train forward + VJP(c) -> parameter gradient v ----
  k_mm_bias_tanh<<<(NTR/16)*(HID/16), W, 0, stream>>>(Xtr, W1, b1, BUF1, NTR, D_IN, HID);
  k_mm_bias_tanh<<<(NTR/16)*(HID/16), W, 0, stream>>>(BUF1, W2, b2, BUF2, NTR, HID, HID);
  k_gw3<<<(HID*DOUT + 255)/256, 256, 0, stream>>>(BUF2, Ctr, gW3);
  k_colsum<<<1, 32, 0, stream>>>(Ctr, gb3, NTR, DOUT);
  k_dz2<<<(NTR*HID + 255)/256, 256, 0, stream>>>(Ctr, W3, BUF2, BUF3);
  k_mm_at<<<(HID/16)*(HID/16), W, 0, stream>>>(BUF1, BUF3, gW2, HID, NTR, HID);  // gW2 = H1^T @ dZ2
  k_colsum<<<1, 256, 0, stream>>>(BUF3, gb2, NTR, HID);
  k_mm_bt_dtanh<<<(NTR/16)*(HID/16), W, 0, stream>>>(BUF3, W2, BUF1, BUF2, NTR, HID, HID); // dZ1
  k_mm_at<<<(D_IN/16)*(HID/16), W, 0, stream>>>(Xtr, BUF2, gW1, D_IN, NTR, HID); // gW1 = X^T @ dZ1
  k_colsum<<<1, 256, 0, stream>>>(BUF2, gb1, NTR, HID);

  // ---- Phase B: test forward + JVP(v) -> output ----
  k_test_l1<<<(NTE/16)*(HID/16), W, 0, stream>>>(Xte, W1, b1, gW1, gb1, BUF1, BUF3);
  k_test_l2<<<(NTE/16)*(HID/16), W, 0, stream>>>(BUF1, BUF3, W2, b2, gW2, gb2, BUF2, BUF4);
  k_test_out<<<(NTE*DOUT + 255)/256, 256, 0, stream>>>(BUF2, BUF4, W3, gW3, gb3, out);
}